// HyperNet_30511447671260
// MI455X (gfx1250) — compile-verified
//
#include <hip/hip_runtime.h>
#include <math.h>

typedef __attribute__((ext_vector_type(2))) float v2f;
typedef __attribute__((ext_vector_type(8))) float v8f;

#define DIMW 32
#define DD 31
#define NW 256
#define QPAD 34   // LDS row stride: bank-conflict-free and 8B-aligned rows
// ws layout per batch (floats): [0:32) th, [32:1056) Qt TRANSPOSED (32x32: W[32+c*32+d]=Q[d][c]),
// [1056:1088) ro_tail (padded), [1088] zeta_t, [1089] coef_t, [1090] tanh(bias_head[u])
#define WS_STRIDE 1120

__device__ __forceinline__ float gelu_exact(float x) {
    return 0.5f * x * (1.0f + erff(x * 0.70710678118654752f));
}

// One block per batch row. Wave 0: head chain; Wave 1: tail chain.
__global__ void build_params_kernel(const int* __restrict__ u,
                                    const int* __restrict__ r,
                                    const float* __restrict__ emb,
                                    const float* __restrict__ bias_head,
                                    const float* __restrict__ head_rot_w,
                                    const float* __restrict__ head_boost_w,
                                    const float* __restrict__ tail_rot_w,
                                    const float* __restrict__ tail_boost_w,
                                    float* __restrict__ ws) {
    __shared__ float Wb[2][DD][DD + 2];   // gelu(rot_w) rows, padded stride 33
    __shared__ float Sb[2][DD];           // 2 / ||w_i||^2
    __shared__ float Qs[DIMW][DIMW + 1];  // head Q staged (padded 32x32)
    __shared__ float Rb[2][DIMW];         // ro = tanh(boost_w)/32 (padded)
    __shared__ float red[2];              // v2 per wave
    __shared__ float prodbuf[DIMW];       // head dot scratch
    __shared__ float dred;                // head dot
    __shared__ float hb[DIMW];            // head embedding

    const int b    = blockIdx.x;
    const int ridx = r[b];
    const int ui   = u[b];
    const int wave = threadIdx.x >> 5;
    const int lane = threadIdx.x & 31;

    const float* rotw = (wave == 0 ? head_rot_w : tail_rot_w) + (size_t)ridx * (DD * DD);
    const float* bstw = (wave == 0 ? head_boost_w : tail_boost_w) + (size_t)ridx * DD;

    // Stage gelu rows + per-row inverse norms + boost vector
    if (lane < DD) {
        float ss = 0.f;
        #pragma unroll 1
        for (int k = 0; k < DD; ++k) {
            float wv = gelu_exact(rotw[lane * DD + k]);
            Wb[wave][lane][k] = wv;
            ss += wv * wv;
        }
        Sb[wave][lane] = 2.0f / ss;
        Rb[wave][lane] = tanhf(bstw[lane]) * (1.0f / (float)DIMW);
    }
    if (lane == 31) Rb[wave][31] = 0.f;
    if (threadIdx.x < DIMW) hb[threadIdx.x] = emb[(size_t)ui * DIMW + threadIdx.x];
    __syncthreads();

    // Householder chain: lane j owns row j of C; C <- C - s*(C w) w^T, i ascending.
    float c[DD];
    #pragma unroll
    for (int k = 0; k < DD; ++k) c[k] = (lane == k) ? 1.f : 0.f;
    if (lane < DD) {
        for (int i = 0; i < DD; ++i) {
            float uacc = 0.f;
            #pragma unroll
            for (int k = 0; k < DD; ++k) uacc += c[k] * Wb[wave][i][k];
            uacc *= Sb[wave][i];
            #pragma unroll
            for (int k = 0; k < DD; ++k) c[k] -= uacc * Wb[wave][i][k];
        }
    }
    if (lane == 0) {
        float v2 = 0.f;
        for (int k = 0; k < DD; ++k) v2 += Rb[wave][k] * Rb[wave][k];
        red[wave] = v2;
    }
    __syncthreads();

    const float v2   = red[wave];
    const float zeta = 1.0f / (sqrtf(1.0f - v2) + 1e-8f);
    const float coef = (zeta - 1.0f) / (v2 + 1e-9f);

    float* W = ws + (size_t)b * WS_STRIDE;
    if (wave == 1) {
        // Store Q_tail TRANSPOSED: W[32 + c*32 + d] = Q[d][c] (lane-coalesced per k).
        if (lane < DD) {
            for (int k = 0; k < DD; ++k) W[32 + k * 32 + lane] = c[k];
        } else {
            for (int k = 0; k < DIMW; ++k) W[32 + k * 32 + 31] = 0.f;  // Q[31][c] = 0
            for (int d = 0; d < DD; ++d)  W[32 + 31 * 32 + d] = 0.f;   // Q[d][31] = 0
        }
        W[1056 + lane] = Rb[1][lane];
        if (lane == 0) {
            W[1088] = zeta;
            W[1089] = coef;
            W[1090] = tanhf(bias_head[ui]);
        }
    } else {
        // Stage Q_head into LDS (padded, row-major: Qs[d][c])
        if (lane < DD) {
            for (int k = 0; k < DD; ++k) Qs[lane][k] = c[k];
            Qs[lane][31] = 0.f;
        } else {
            for (int k = 0; k < 32; ++k) Qs[31][k] = 0.f;
        }
    }
    __syncthreads();

    // Head finish: x_rot[c] = sum_d h[1+d] * Qh[d][c], then boost.
    float xr = 0.f;
    if (wave == 0) {
        if (lane < DD) {
            #pragma unroll 1
            for (int d = 0; d < DD; ++d) xr += hb[1 + d] * Qs[d][lane];
            prodbuf[lane] = xr * Rb[0][lane];
        } else {
            prodbuf[31] = 0.f;
        }
    }
    __syncthreads();
    if (wave == 0 && lane == 0) {
        float dt = 0.f;
        for (int k = 0; k < DD; ++k) dt += prodbuf[k];
        dred = dt;
    }
    __syncthreads();
    if (wave == 0) {
        const float dot = dred;
        const float t0  = hb[0];
        if (lane == 0) W[0] = zeta * t0 - zeta * dot;
        if (lane < DD) {
            float roc = Rb[0][lane];
            W[1 + lane] = -zeta * t0 * roc + xr + coef * roc * dot;
        }
    }
}

// One block (256 threads, 8 waves) per batch row. WMMA rotation + boost + distance.
__global__ void tail_kernel(const int* __restrict__ v,
                            const float* __restrict__ emb,
                            const float* __restrict__ bias_tail,
                            const float* __restrict__ ws,
                            float* __restrict__ out) {
    __shared__ float X[NW][QPAD];     // A matrix (then reused for rotated result)
    __shared__ float T0[NW];
    __shared__ float Qs[DIMW][QPAD];  // Qs[c][d] = Q[d][c] (transposed in ws)
    __shared__ float ro_s[DIMW];
    __shared__ float th_s[DIMW];
    __shared__ float sc[4];

    const int b   = blockIdx.x;
    const int tid = threadIdx.x;
    const float* W = ws + (size_t)b * WS_STRIDE;

    const int vi = v[(size_t)b * NW + tid];
    const float* e = emb + (size_t)vi * DIMW;
    T0[tid] = e[0];
    #pragma unroll 1
    for (int d = 0; d < DD; ++d) X[tid][d] = e[1 + d];
    X[tid][31] = 0.f;
    if (tid < DIMW) {
        for (int k = 0; k < DIMW; ++k) Qs[tid][k] = W[32 + tid * 32 + k];
        ro_s[tid] = W[1056 + tid];
        th_s[tid] = W[tid];
    }
    if (tid == 0) { sc[0] = W[1088]; sc[1] = W[1089]; sc[2] = W[1090]; }
    __syncthreads();

    const int wv   = tid >> 5;
    const int lane = tid & 31;
    const int half = lane >> 4;
    const int l16  = lane & 15;

    // Per-wave B fragments for both N-tiles: contiguous (k0, k0+1) pairs from
    // the transposed Qs -> adjacent dest VGPRs, no repacking movs.
    v2f bfr0[8], bfr1[8];
    #pragma unroll
    for (int kk = 0; kk < 8; ++kk) {
        const int k0 = 4 * kk + 2 * half;
        bfr0[kk].x = Qs[l16][k0];        bfr0[kk].y = Qs[l16][k0 + 1];
        bfr1[kk].x = Qs[16 + l16][k0];   bfr1[kk].y = Qs[16 + l16][k0 + 1];
    }

    #pragma unroll
    for (int t2 = 0; t2 < 2; ++t2) {
        const int mt = wv + 8 * t2;     // 16 M-tiles over 8 waves
        const int mr = mt * 16 + l16;
        v8f acc0 = {0.f, 0.f, 0.f, 0.f, 0.f, 0.f, 0.f, 0.f};
        v8f acc1 = {0.f, 0.f, 0.f, 0.f, 0.f, 0.f, 0.f, 0.f};
        #pragma unroll
        for (int kk = 0; kk < 8; ++kk) {
            const int k0 = 4 * kk + 2 * half;
            v2f a;
            a.x = X[mr][k0];
            a.y = X[mr][k0 + 1];
            acc0 = __builtin_amdgcn_wmma_f32_16x16x4_f32(false, a, false, bfr0[kk],
                                                         (short)0, acc0, false, false);
            acc1 = __builtin_amdgcn_wmma_f32_16x16x4_f32(false, a, false, bfr1[kk],
                                                         (short)0, acc1, false, false);
        }
        // Write rotated rows back in place (this wave's rows only; reads are done).
        #pragma unroll
        for (int i = 0; i < 8; ++i) {
            const int row = mt * 16 + i + 8 * half;
            X[row][l16]      = acc0[i];
            X[row][16 + l16] = acc1[i];
        }
    }
    __syncthreads();

    // Boost + Lorentz distance per row
    const float zeta = sc[0], coef = sc[1], bh = sc[2];
    const float t0 = T0[tid];
    float dot = 0.f;
    #pragma unroll 1
    for (int c2 = 0; c2 < DD; ++c2) dot += X[tid][c2] * ro_s[c2];
    const float tt0 = zeta * t0 - zeta * dot;
    const float d0  = tt0 - th_s[0];
    float acc = -d0 * d0;   // mkv = sum(d*d) - 2*d0^2  ->  sum_{c} dc^2 - d0^2
    #pragma unroll 1
    for (int c2 = 0; c2 < DD; ++c2) {
        float roc = ro_s[c2];
        float ttc = -zeta * t0 * roc + X[tid][c2] + coef * roc * dot;
        float dc  = ttc - th_s[1 + c2];
        acc += dc * dc;
    }
    const float bt = tanhf(bias_tail[vi]);
    out[(size_t)b * NW + tid] = 0.85f - acc + bh + bt;
}

extern "C" void kernel_launch(void* const* d_in, const int* in_sizes, int n_in,
                              void* d_out, int out_size, void* d_ws, size_t ws_size,
                              hipStream_t stream) {
    const int*   u         = (const int*)d_in[0];
    const int*   r         = (const int*)d_in[1];
    const int*   v         = (const int*)d_in[2];
    const float* emb       = (const float*)d_in[3];
    const float* bias_head = (const float*)d_in[4];
    const float* bias_tail = (const float*)d_in[5];
    const float* hrw       = (const float*)d_in[6];
    const float* hbw       = (const float*)d_in[7];
    const float* trw       = (const float*)d_in[8];
    const float* tbw       = (const float*)d_in[9];
    float* out = (float*)d_out;
    float* ws  = (float*)d_ws;

    const int B = in_sizes[0];
    const int N = in_sizes[2] / B;   // 256 per reference
    (void)N; (void)out_size; (void)ws_size; (void)n_in;

    build_params_kernel<<<B, 64, 0, stream>>>(u, r, emb, bias_head, hrw, hbw, trw, tbw, ws);
    tail_kernel<<<B, NW, 0, stream>>>(v, emb, bias_tail, ws, out);
}